// RWKV_RNN_60868276519710
// MI455X (gfx1250) — compile-verified
//
#include <hip/hip_runtime.h>
#include <hip/hip_bf16.h>
#include <math.h>

// ---------------------------------------------------------------------------
// RWKV v4 single-token inference for gfx1250 (MI455X).
// Bandwidth-bound matvec chain; matvecs via V_WMMA_F32_16X16X4_F32 with
// per-block 16-row strips and K split across the 8 waves (deterministic
// LDS combine). Weight streams are non-temporal.
// ---------------------------------------------------------------------------

typedef __attribute__((ext_vector_type(2))) float v2f;
typedef __attribute__((ext_vector_type(8))) float v8f;

#define DD   1024
#define HH   4096
#define NL   12
#define NV   50277

// ---------------- block-wide reduction (256 threads = 8 waves of 32) -------
__device__ __forceinline__ float block_reduce_sum(float v, float* red) {
  __syncthreads();                       // protect 'red' reuse across calls
  int lane = threadIdx.x & 31;
  int wid  = threadIdx.x >> 5;
#pragma unroll
  for (int o = 16; o > 0; o >>= 1) v += __shfl_xor(v, o, 32);
  if (lane == 0) red[wid] = v;
  __syncthreads();
  if (threadIdx.x == 0) {
    float s = 0.f;
#pragma unroll
    for (int i = 0; i < 8; ++i) s += red[i];
    red[0] = s;
  }
  __syncthreads();
  return red[0];
}

// LN stats over D=1024 (4 strided elements / thread), returns mu & rsqrt(var+eps)
__device__ __forceinline__ void ln_stats(const float* __restrict__ x,
                                         float* red, float lx[4],
                                         float& mu, float& rs) {
  int t = threadIdx.x;
  float s = 0.f;
#pragma unroll
  for (int j = 0; j < 4; ++j) { lx[j] = x[t + 256 * j]; s += lx[j]; }
  s = block_reduce_sum(s, red);
  mu = s * (1.0f / 1024.0f);
  float v = 0.f;
#pragma unroll
  for (int j = 0; j < 4; ++j) { float d = lx[j] - mu; v += d * d; }
  v = block_reduce_sum(v, red);
  rs = rsqrtf(v * (1.0f / 1024.0f) + 1e-5f);
}

__device__ __forceinline__ float sigmoidf_(float x) {
  return 1.0f / (1.0f + expf(-x));
}

// ---------------- WMMA matvec core -----------------------------------------
// One 256-thread block computes y[rowBase .. rowBase+15] = W[rows x K] @ xs[K].
// Each of the 8 waves handles a K/8 slice with V_WMMA_F32_16X16X4_F32 (x in
// column 0 of B); the 8 partial accumulators are combined in fixed order via
// LDS. Returns the final dot product for threads t<16 (row = rowBase+t).
// A layout (16x4 f32): lane holds M=lane&15, K-pair selected by lane>>4.
// Result column N=0 lives in lane 0 (M=0..7 in c[0..7]) and lane 16 (M=8..15).
__device__ __forceinline__ float block_matvec16(const float* __restrict__ W,
                                                const float* __restrict__ xs,
                                                int K, int nrows, int rowBase,
                                                float* __restrict__ accLds) {
  const int t    = threadIdx.x;
  const int lane = t & 31;
  const int wv   = t >> 5;                 // wave id 0..7
  const int half = lane >> 4;              // 0 or 1
  int mrow = rowBase + (lane & 15);
  if (mrow >= nrows) mrow = nrows - 1;     // clamp tail rows (stores guarded)
  const float* Wr = W + (size_t)mrow * (size_t)K;
  const int kslice = K >> 3;               // per-wave K slice
  const int kbeg   = wv * kslice;
  const int kend   = kbeg + kslice;
  const int koff   = half << 1;            // 0 or 2
  const float msk  = ((lane & 15) == 0) ? 1.f : 0.f;  // only column 0 carries x
  v8f c = (v8f){0.f, 0.f, 0.f, 0.f, 0.f, 0.f, 0.f, 0.f};
#pragma unroll 8
  for (int k0 = kbeg; k0 < kend; k0 += 4) {
    v2f a = __builtin_nontemporal_load((const v2f*)(Wr + k0 + koff));  // stream
    v2f bv = *(const v2f*)(xs + k0 + koff);   // LDS broadcast, unconditional
    v2f b;
    b.x = bv.x * msk;
    b.y = bv.y * msk;
    c = __builtin_amdgcn_wmma_f32_16x16x4_f32(false, a, false, b,
                                              (short)0, c, false, false);
  }
  if ((lane & 15) == 0) {
#pragma unroll
    for (int j = 0; j < 8; ++j) accLds[wv * 16 + half * 8 + j] = c[j];
  }
  __syncthreads();
  float r = 0.f;
  if (t < 16) {
#pragma unroll
    for (int w = 0; w < 8; ++w) r += accLds[w * 16 + t];  // fixed order
  }
  return r;
}

// ---------------- kernels ----------------------------------------------------

// x = LN0(emb[token])
__global__ void k_embed(const int* __restrict__ token,
                        const float* __restrict__ emb,
                        const float* __restrict__ w, const float* __restrict__ b,
                        float* __restrict__ X) {
  __shared__ float red[8];
  int t = threadIdx.x;
  const float* e = emb + (size_t)token[0] * DD;
  float lx[4], mu, rs;
  ln_stats(e, red, lx, mu, rs);
#pragma unroll
  for (int j = 0; j < 4; ++j) {
    int i = t + 256 * j;
    X[i] = (lx[j] - mu) * rs * w[i] + b[i];
  }
}

// XLN = LN(X, lnout_w, lnout_b)   (1 block)
__global__ void k_lnout(const float* __restrict__ X,
                        const float* __restrict__ w, const float* __restrict__ b,
                        float* __restrict__ XLN) {
  __shared__ float red[8];
  int t = threadIdx.x;
  float lx[4], mu, rs;
  ln_stats(X, red, lx, mu, rs);
#pragma unroll
  for (int j = 0; j < 4; ++j) {
    int i = t + 256 * j;
    XLN[i] = (lx[j] - mu) * rs * w[i] + b[i];
  }
}

// LN1 + token-mix prologue, then {kw,vw,rw} matvec selected by blockIdx.y
__global__ void k_att_qkv(const float* __restrict__ X,
                          const float* __restrict__ s_sa,
                          float* __restrict__ xx_out,
                          const float* __restrict__ l1w, const float* __restrict__ l1b,
                          const float* __restrict__ tmk, const float* __restrict__ tmv,
                          const float* __restrict__ tmr,
                          const float* __restrict__ kw, const float* __restrict__ vw,
                          const float* __restrict__ rw,
                          float* __restrict__ ok, float* __restrict__ ov,
                          float* __restrict__ orr) {
  __shared__ float xs[DD];
  __shared__ float red[8];
  __shared__ float acc[128];
  int t = threadIdx.x;
  int sel = blockIdx.y;
  float lx[4], mu, rs;
  ln_stats(X, red, lx, mu, rs);
  const float* tm = (sel == 0) ? tmk : (sel == 1) ? tmv : tmr;
#pragma unroll
  for (int j = 0; j < 4; ++j) {
    int i = t + 256 * j;
    float xxv = (lx[j] - mu) * rs * l1w[i] + l1b[i];
    float m = tm[i];
    xs[i] = xxv * m + s_sa[i] * (1.0f - m);
    if (blockIdx.x == 0 && sel == 0) xx_out[i] = xxv;  // new state row 5l+1
  }
  __syncthreads();
  const float* W = (sel == 0) ? kw : (sel == 1) ? vw : rw;
  float* out     = (sel == 0) ? ok : (sel == 1) ? ov : orr;
  int rowBase = blockIdx.x * 16;
  float r = block_matvec16(W, xs, DD, DD, rowBase, acc);
  if (t < 16) out[rowBase + t] = r;
}

// wkv + state update prologue, then x += ow @ (r*wkv)
__global__ void k_att_wkv(float* __restrict__ X,
                          const float* __restrict__ st,   // in_state + 5l*D
                          float* __restrict__ ost,        // out_state + 5l*D
                          const float* __restrict__ tf, const float* __restrict__ td,
                          const float* __restrict__ kv, const float* __restrict__ vv,
                          const float* __restrict__ rp,
                          const float* __restrict__ ow) {
  __shared__ float xs[DD];
  __shared__ float acc[128];
  int t = threadIdx.x;
#pragma unroll
  for (int j = 0; j < 4; ++j) {
    int i = t + 256 * j;
    float aa = st[2 * DD + i], bb = st[3 * DD + i], pp = st[4 * DD + i];
    float kk = kv[i], v1 = vv[i];
    float ww = tf[i] + kk;
    float p  = fmaxf(pp, ww);
    float e1 = expf(pp - p), e2 = expf(ww - p);
    float wkv = (e1 * aa + e2 * v1) / (e1 * bb + e2);
    xs[i] = sigmoidf_(rp[i]) * wkv;
    float ww2 = pp + td[i];
    float p2  = fmaxf(ww2, kk);
    float f1  = expf(ww2 - p2), f2 = expf(kk - p2);
    if (blockIdx.x == 0) {
      ost[2 * DD + i] = f1 * aa + f2 * v1;   // aa2
      ost[3 * DD + i] = f1 * bb + f2;        // bb2
      ost[4 * DD + i] = p2;                  // p2
    }
  }
  __syncthreads();
  int rowBase = blockIdx.x * 16;
  float r = block_matvec16(ow, xs, DD, DD, rowBase, acc);
  if (t < 16) X[rowBase + t] += r;
}

// LN2 + channel-mix prologue; grid.y=0: fk = relu(fkw@fxk)^2 ; grid.y=1: frpre
__global__ void k_ffn_kr(const float* __restrict__ X,
                         const float* __restrict__ s_ff,
                         float* __restrict__ yy_out,
                         const float* __restrict__ l2w, const float* __restrict__ l2b,
                         const float* __restrict__ fmk, const float* __restrict__ fmr,
                         const float* __restrict__ fkw, const float* __restrict__ frw,
                         float* __restrict__ fk, float* __restrict__ frp) {
  int sel = blockIdx.y;
  if (sel == 1 && blockIdx.x >= 64) return;   // uniform early-out
  __shared__ float xs[DD];
  __shared__ float red[8];
  __shared__ float acc[128];
  int t = threadIdx.x;
  float lx[4], mu, rs;
  ln_stats(X, red, lx, mu, rs);
  const float* tm = sel ? fmr : fmk;
#pragma unroll
  for (int j = 0; j < 4; ++j) {
    int i = t + 256 * j;
    float yv = (lx[j] - mu) * rs * l2w[i] + l2b[i];
    float m = tm[i];
    xs[i] = yv * m + s_ff[i] * (1.0f - m);
    if (blockIdx.x == 0 && sel == 0) yy_out[i] = yv;   // new state row 5l+0
  }
  __syncthreads();
  int rowBase = blockIdx.x * 16;
  if (sel == 0) {
    float r = block_matvec16(fkw, xs, DD, HH, rowBase, acc);
    if (t < 16) { float z = fmaxf(r, 0.f); fk[rowBase + t] = z * z; }
  } else {
    float r = block_matvec16(frw, xs, DD, DD, rowBase, acc);
    if (t < 16) frp[rowBase + t] = r;
  }
}

// x += sigmoid(frpre) * (fvw @ fk)   (K = 4096)
__global__ void k_ffn_out(float* __restrict__ X,
                          const float* __restrict__ fk,
                          const float* __restrict__ frp,
                          const float* __restrict__ fvw) {
  __shared__ float xs[HH];                 // 16 KB
  __shared__ float acc[128];
  int t = threadIdx.x;
#pragma unroll
  for (int j = 0; j < 16; ++j) xs[t + 256 * j] = fk[t + 256 * j];
  __syncthreads();
  int rowBase = blockIdx.x * 16;
  float r = block_matvec16(fvw, xs, HH, DD, rowBase, acc);
  if (t < 16) {
    int m = rowBase + t;
    X[m] += sigmoidf_(frp[m]) * r;
  }
}

// logits = head @ XLN   (50277 rows, tail-guarded)
__global__ void k_head(const float* __restrict__ XLN,
                       const float* __restrict__ head, float* __restrict__ logits) {
  __shared__ float xs[DD];
  __shared__ float acc[128];
  int t = threadIdx.x;
#pragma unroll
  for (int j = 0; j < 4; ++j) xs[t + 256 * j] = XLN[t + 256 * j];
  __syncthreads();
  int rowBase = blockIdx.x * 16;
  float r = block_matvec16(head, xs, DD, NV, rowBase, acc);
  if (t < 16) {
    int m = rowBase + t;
    if (m < NV) logits[m] = r;
  }
}

// ---------------- host launch -----------------------------------------------
extern "C" void kernel_launch(void* const* d_in, const int* in_sizes, int n_in,
                              void* d_out, int out_size, void* d_ws, size_t ws_size,
                              hipStream_t stream) {
  (void)in_sizes; (void)n_in; (void)out_size; (void)ws_size;
  const int*   token  = (const int*)  d_in[0];
  const float* state  = (const float*)d_in[1];
  const float* emb    = (const float*)d_in[2];
  const float* ln0_w  = (const float*)d_in[3];
  const float* ln0_b  = (const float*)d_in[4];
  const float* ln1_w  = (const float*)d_in[5];
  const float* ln1_b  = (const float*)d_in[6];
  const float* tmk    = (const float*)d_in[7];
  const float* tmv    = (const float*)d_in[8];
  const float* tmr    = (const float*)d_in[9];
  const float* tf     = (const float*)d_in[10];
  const float* td     = (const float*)d_in[11];
  const float* kw     = (const float*)d_in[12];
  const float* vw     = (const float*)d_in[13];
  const float* rw     = (const float*)d_in[14];
  const float* ow     = (const float*)d_in[15];
  const float* ln2_w  = (const float*)d_in[16];
  const float* ln2_b  = (const float*)d_in[17];
  const float* fmk    = (const float*)d_in[18];
  const float* fmr    = (const float*)d_in[19];
  const float* fkw    = (const float*)d_in[20];
  const float* fvw    = (const float*)d_in[21];
  const float* frw    = (const float*)d_in[22];
  const float* lno_w  = (const float*)d_in[23];
  const float* lno_b  = (const float*)d_in[24];
  const float* head   = (const float*)d_in[25];

  float* logits = (float*)d_out;          // [V]
  float* nstate = logits + NV;            // [5L*D]

  float* wsf = (float*)d_ws;              // 40 KB of scratch
  float* X   = wsf;                       // [D]
  float* Kv  = wsf + 1024;                // k
  float* Vv  = wsf + 2048;                // v
  float* Rp  = wsf + 3072;                // r (pre-sigmoid)
  float* FRp = wsf + 4096;                // frw@fxr (pre-sigmoid)
  float* XLN = wsf + 5120;                // final LN(x)
  float* FK  = wsf + 6144;                // relu(fkw@fxk)^2  [H]

  k_embed<<<1, 256, 0, stream>>>(token, emb, ln0_w, ln0_b, X);

  for (int l = 0; l < NL; ++l) {
    const float* l1w_ = ln1_w + l * DD;  const float* l1b_ = ln1_b + l * DD;
    const float* l2w_ = ln2_w + l * DD;  const float* l2b_ = ln2_b + l * DD;
    const float* tmk_ = tmk + l * DD;    const float* tmv_ = tmv + l * DD;
    const float* tmr_ = tmr + l * DD;
    const float* tf_  = tf  + l * DD;    const float* td_  = td  + l * DD;
    const float* fmk_ = fmk + l * DD;    const float* fmr_ = fmr + l * DD;
    const float* kw_  = kw  + (size_t)l * DD * DD;
    const float* vw_  = vw  + (size_t)l * DD * DD;
    const float* rw_  = rw  + (size_t)l * DD * DD;
    const float* ow_  = ow  + (size_t)l * DD * DD;
    const float* frw_ = frw + (size_t)l * DD * DD;
    const float* fkw_ = fkw + (size_t)l * HH * DD;
    const float* fvw_ = fvw + (size_t)l * DD * HH;
    const float* st_  = state  + (size_t)(5 * l) * DD;
    float*       ost_ = nstate + (size_t)(5 * l) * DD;

    k_att_qkv<<<dim3(64, 3), 256, 0, stream>>>(X, st_ + DD, ost_ + DD,
                                               l1w_, l1b_, tmk_, tmv_, tmr_,
                                               kw_, vw_, rw_, Kv, Vv, Rp);
    k_att_wkv<<<64, 256, 0, stream>>>(X, st_, ost_, tf_, td_, Kv, Vv, Rp, ow_);
    k_ffn_kr<<<dim3(256, 2), 256, 0, stream>>>(X, st_, ost_, l2w_, l2b_,
                                               fmk_, fmr_, fkw_, frw_, FK, FRp);
    k_ffn_out<<<64, 256, 0, stream>>>(X, FK, FRp, fvw_);
  }

  k_lnout<<<1, 256, 0, stream>>>(X, lno_w, lno_b, XLN);
  k_head<<<(NV + 15) / 16, 256, 0, stream>>>(XLN, head, logits);
}